// Head_46325517254812
// MI455X (gfx1250) — compile-verified
//
#include <hip/hip_runtime.h>
#include <hip/hip_bf16.h>

typedef __attribute__((ext_vector_type(2))) float v2f;
typedef __attribute__((ext_vector_type(8))) float v8f;

#define NUM_CLASSES 80
#define NIMG 16
#define NG   512
#define NA   3
#define PAIRS (NA * NG * 5)          // 7680
#define PAIR_BLOCKS 30               // 30 * 256 == 7680
#define OBJ_BLOCKS 120

// Fast BCE-with-logits: max(x,0) - x*t + log1p(exp(-|x|)).
// __expf/__logf lower to single v_exp_f32/v_log_f32 TRANS ops; abs error of
// log(1+e) vs log1p is <=~1e-7, negligible in a mean of O(1) terms.
__device__ __forceinline__ float bce_logits(float x, float t) {
    return fmaxf(x, 0.0f) - x * t + __logf(1.0f + __expf(-fabsf(x)));
}
__device__ __forceinline__ float sigmoidf(float x) {
    return 1.0f / (1.0f + __expf(-x));
}

// -------------------------------------------------------------------------
// Pair kernel: one thread per (anchor, gt, offset) triple.
// Produces per-block partial sums of {box_loss_num, pm_count, cls_loss_num}
// and scatters packed (pair_idx, obj_val) into gt_object via atomicMax so the
// duplicate-cell resolution is deterministic (last pair index wins).
// -------------------------------------------------------------------------
__global__ void yolo_pair_kernel(const float* __restrict__ pred,
                                 const float* __restrict__ gt_boxes,   // [G,4] xyxy
                                 const float* __restrict__ anchors,    // [3,3,2]
                                 const int*   __restrict__ gt_labels,  // [G]
                                 const int*   __restrict__ image_ids,  // [G]
                                 unsigned long long* __restrict__ gtobj, // [N*H*W*A]
                                 float* __restrict__ partials,         // [gridDim.x*3]
                                 int H, int W, float stride, int level)
{
    const int t = blockIdx.x * blockDim.x + threadIdx.x;   // 0..7679 exactly
    float box_c = 0.0f, pm_c = 0.0f, cls_c = 0.0f;

    const int a = t / (NG * 5);
    const int g = (t / 5) % NG;
    const int o = t % 5;

    // gt box -> cxcywh
    const float x0 = gt_boxes[g * 4 + 0], y0 = gt_boxes[g * 4 + 1];
    const float x1 = gt_boxes[g * 4 + 2], y1 = gt_boxes[g * 4 + 3];
    const float gcx = 0.5f * (x0 + x1), gcy = 0.5f * (y0 + y1);
    const float gw = x1 - x0, gh = y1 - y0;

    const float aw = anchors[(level * NA + a) * 2 + 0];
    const float ah = anchors[(level * NA + a) * 2 + 1];

    // size match test
    const float rw = aw / gw, rh = ah / gh;
    const float mr = fmaxf(fmaxf(rw, 1.0f / rw), fmaxf(rh, 1.0f / rh));
    const bool match = mr < 4.0f;

    // grid-space center, neighbor validity
    const float xg = gcx / stride, yg = gcy / stride;
    const float fxx = xg - floorf(xg), fxy = yg - floorf(yg);
    const float invx = (float)W - xg, invy = (float)H - yg;

    bool valid;
    if      (o == 0) valid = true;
    else if (o == 1) valid = (fxx < 0.5f) && (xg > 1.0f);
    else if (o == 2) valid = (fxy < 0.5f) && (yg > 1.0f);
    else if (o == 3) valid = ((invx - floorf(invx)) < 0.5f) && (invx > 1.0f);
    else             valid = ((invy - floorf(invy)) < 0.5f) && (invy > 1.0f);

    const float offx_t[5] = {0.0f, 0.5f, 0.0f, -0.5f, 0.0f};
    const float offy_t[5] = {0.0f, 0.0f, 0.5f, 0.0f, -0.5f};
    const int gridx = (int)floorf(xg - offx_t[o]);
    const int gridy = (int)floorf(yg - offy_t[o]);
    const int gx = min(max(gridx, 0), W - 1);
    const int gy = min(max(gridy, 0), H - 1);

    if (match && valid) {
        const int img = image_ids[g];
        const float* p = pred + ((((size_t)img * H + gy) * W + gx) * NA + a) * (5 + NUM_CLASSES);

        const float bx = 2.0f * sigmoidf(p[0]) - 0.5f + (float)gridx;  // unclipped cell
        const float by = 2.0f * sigmoidf(p[1]) - 0.5f + (float)gridy;
        const float sw = sigmoidf(p[2]), sh = sigmoidf(p[3]);
        const float bw = 4.0f * sw * sw * aw / stride;
        const float bh = 4.0f * sh * sh * ah / stride;

        // gt in grid units
        const float tx = gcx / stride, ty = gcy / stride;
        const float tw = gw / stride,  th = gh / stride;

        // GIoU (cxcywh)
        const float b1x0 = bx - bw * 0.5f, b1x1 = bx + bw * 0.5f;
        const float b1y0 = by - bh * 0.5f, b1y1 = by + bh * 0.5f;
        const float b2x0 = tx - tw * 0.5f, b2x1 = tx + tw * 0.5f;
        const float b2y0 = ty - th * 0.5f, b2y1 = ty + th * 0.5f;
        const float iw = fmaxf(fminf(b1x1, b2x1) - fmaxf(b1x0, b2x0), 0.0f);
        const float ih = fmaxf(fminf(b1y1, b2y1) - fmaxf(b1y0, b2y0), 0.0f);
        const float inter = iw * ih;
        const float uni = bw * bh + tw * th - inter;
        const float cw = fmaxf(b1x1, b2x1) - fminf(b1x0, b2x0);
        const float ch = fmaxf(b1y1, b2y1) - fminf(b1y0, b2y0);
        const float carea = cw * ch;
        const float iou  = inter / (uni + 1e-7f);
        const float giou = iou - (carea - uni) / (carea + 1e-7f);

        box_c = 1.0f - giou;
        pm_c  = 1.0f;

        // deterministic last-pair-wins scatter of clip(giou, 0, inf)
        const float obj_val = fmaxf(giou, 0.0f);
        const size_t cell = (((size_t)img * H + gy) * W + gx) * NA + a;
        const unsigned long long pk =
            ((unsigned long long)(unsigned)(t + 1) << 32) |
            (unsigned long long)__float_as_uint(obj_val);
        atomicMax(&gtobj[cell], pk);

        // class BCE over 80 classes vs one-hot(label)
        const int label = gt_labels[g];
        float cs = 0.0f;
        #pragma unroll 4
        for (int c = 0; c < NUM_CLASSES; ++c) {
            const float tt = (c == label) ? 1.0f : 0.0f;
            cs += bce_logits(p[5 + c], tt);
        }
        cls_c = cs;
    }

    // fixed-order block reduction (shuffle within wave32, LDS across 8 waves)
    #pragma unroll
    for (int m = 16; m >= 1; m >>= 1) {
        box_c += __shfl_xor(box_c, m, 32);
        pm_c  += __shfl_xor(pm_c,  m, 32);
        cls_c += __shfl_xor(cls_c, m, 32);
    }
    __shared__ float sb[8], sp[8], sc[8];
    const int wave = threadIdx.x >> 5, lane = threadIdx.x & 31;
    if (lane == 0) { sb[wave] = box_c; sp[wave] = pm_c; sc[wave] = cls_c; }
    __syncthreads();
    if (threadIdx.x == 0) {
        float b = 0.0f, pcnt = 0.0f, c = 0.0f;
        #pragma unroll
        for (int i = 0; i < 8; ++i) { b += sb[i]; pcnt += sp[i]; c += sc[i]; }
        partials[blockIdx.x * 3 + 0] = b;
        partials[blockIdx.x * 3 + 1] = pcnt;
        partials[blockIdx.x * 3 + 2] = c;
    }
}

// -------------------------------------------------------------------------
// Objectness kernel: sum of bce(pred[...,4], gt_object) over all M cells.
// Each wave folds 64 fresh f32 BCE values per step into a 16x16 f32
// accumulator with V_WMMA_F32_16X16X4_F32 using A = ones(16x4):
//   D[m][n] = C[m][n] + sum_k B[k][n]
// Every B element lands in exactly one (k,n), so summing row 0 (lanes 0..15
// of acc[0]) yields the exact total independent of the B lane layout.
// M is a multiple of 64 for all three levels, so the inner loop has no
// bounds guards: EXEC stays all-ones by construction (ISA §7.12 requirement).
// -------------------------------------------------------------------------
__global__ void yolo_obj_kernel(const float* __restrict__ pred,
                                const unsigned long long* __restrict__ gtobj,
                                float* __restrict__ partials,   // [gridDim.x]
                                int M)                          // M % 64 == 0
{
    const int lane = threadIdx.x & 31;
    const int wave_in_blk = threadIdx.x >> 5;
    const int gwave = blockIdx.x * (blockDim.x >> 5) + wave_in_blk;
    const int nwave = gridDim.x * (blockDim.x >> 5);
    const int step = nwave * 64;

    v8f acc = {};
    v2f ones; ones.x = 1.0f; ones.y = 1.0f;

    for (int base = gwave * 64; base < M; base += step) {
        const int i0 = base + lane;
        const int i1 = i0 + 32;
        // prefetch next iteration's strided pred lines into cache
        if (base + step < M) {
            __builtin_prefetch(&pred[(size_t)(i0 + step) * (5 + NUM_CLASSES) + 4], 0, 3);
        }
        const float t0 = __uint_as_float((unsigned)gtobj[i0]);
        const float t1 = __uint_as_float((unsigned)gtobj[i1]);
        const float v0 = bce_logits(pred[(size_t)i0 * (5 + NUM_CLASSES) + 4], t0);
        const float v1 = bce_logits(pred[(size_t)i1 * (5 + NUM_CLASSES) + 4], t1);
        v2f b; b.x = v0; b.y = v1;
        // acc[m][n] += sum_k ones[m][k] * b[k][n]  -> fold 64 values / wmma
        acc = __builtin_amdgcn_wmma_f32_16x16x4_f32(
            false, ones, false, b, (short)0, acc, false, false);
    }

    // row 0 of D lives in acc[0] lanes 0..15 (col sums); sum them.
    float v = acc[0];
    #pragma unroll
    for (int m = 1; m < 16; m <<= 1) v += __shfl_xor(v, m, 16);

    __shared__ float sm[8];
    if (lane == 0) sm[wave_in_blk] = v;
    __syncthreads();
    if (threadIdx.x == 0) {
        float s = 0.0f;
        #pragma unroll
        for (int i = 0; i < 8; ++i) s += sm[i];
        partials[blockIdx.x] = s;
    }
}

// -------------------------------------------------------------------------
// Finalize: serial deterministic combine of all block partials.
// -------------------------------------------------------------------------
__global__ void yolo_finalize_kernel(const float* __restrict__ pairPart, // [3*30*3]
                                     const float* __restrict__ objPart,  // [3*120]
                                     float* __restrict__ out)            // [3]
{
    if (blockIdx.x != 0 || threadIdx.x != 0) return;
    const float Ms[3] = {(float)(NIMG * 80 * 80 * NA),
                         (float)(NIMG * 40 * 40 * NA),
                         (float)(NIMG * 20 * 20 * NA)};
    float lb = 0.0f, lo = 0.0f, lc = 0.0f;
    for (int l = 0; l < 3; ++l) {
        float bs = 0.0f, ps = 0.0f, cs = 0.0f;
        for (int blk = 0; blk < PAIR_BLOCKS; ++blk) {
            bs += pairPart[(l * PAIR_BLOCKS + blk) * 3 + 0];
            ps += pairPart[(l * PAIR_BLOCKS + blk) * 3 + 1];
            cs += pairPart[(l * PAIR_BLOCKS + blk) * 3 + 2];
        }
        const float den = fmaxf(ps, 1.0f);
        lb += bs / den;
        lc += cs / (den * (float)NUM_CLASSES);
        float os = 0.0f;
        for (int blk = 0; blk < OBJ_BLOCKS; ++blk) os += objPart[l * OBJ_BLOCKS + blk];
        lo += os / Ms[l];
    }
    out[0] = 0.05f * lb;   // LW_BOX
    out[1] = 1.00f * lo;   // LW_OBJ
    out[2] = 0.50f * lc;   // LW_CLS
}

extern "C" void kernel_launch(void* const* d_in, const int* in_sizes, int n_in,
                              void* d_out, int out_size, void* d_ws, size_t ws_size,
                              hipStream_t stream) {
    const float* p[3] = {(const float*)d_in[0], (const float*)d_in[1], (const float*)d_in[2]};
    const float* gt_boxes  = (const float*)d_in[3];
    const float* anchors   = (const float*)d_in[4];
    const int*   gt_labels = (const int*)d_in[5];
    const int*   image_ids = (const int*)d_in[6];
    float* out = (float*)d_out;

    const int Hs[3] = {80, 40, 20};
    const int Ws[3] = {80, 40, 20};
    const float strides[3] = {8.0f, 16.0f, 32.0f};
    const int Ms[3] = {NIMG * 80 * 80 * NA, NIMG * 40 * 40 * NA, NIMG * 20 * 20 * NA};

    // workspace layout
    unsigned long long* gtobj[3];
    gtobj[0] = (unsigned long long*)d_ws;
    gtobj[1] = gtobj[0] + Ms[0];
    gtobj[2] = gtobj[1] + Ms[1];
    float* pairPart = (float*)(gtobj[2] + Ms[2]);            // 3*30*3 floats
    float* objPart  = pairPart + 3 * PAIR_BLOCKS * 3;        // 3*120 floats

    const size_t gtobj_bytes = (size_t)(Ms[0] + Ms[1] + Ms[2]) * sizeof(unsigned long long);
    hipMemsetAsync(d_ws, 0, gtobj_bytes, stream);

    for (int l = 0; l < 3; ++l) {
        yolo_pair_kernel<<<PAIR_BLOCKS, 256, 0, stream>>>(
            p[l], gt_boxes, anchors, gt_labels, image_ids,
            gtobj[l], pairPart + l * PAIR_BLOCKS * 3,
            Hs[l], Ws[l], strides[l], l);
    }
    for (int l = 0; l < 3; ++l) {
        yolo_obj_kernel<<<OBJ_BLOCKS, 256, 0, stream>>>(
            p[l], gtobj[l], objPart + l * OBJ_BLOCKS, Ms[l]);
    }
    yolo_finalize_kernel<<<1, 64, 0, stream>>>(pairPart, objPart, out);
}